// Pointnet_plus_28905129902585
// MI455X (gfx1250) — compile-verified
//
#include <hip/hip_runtime.h>
#include <hip/hip_bf16.h>

typedef __attribute__((ext_vector_type(2))) float v2f;
typedef __attribute__((ext_vector_type(8))) float v8f;

#define BATCH 8
#define NSAMP 32
#define EPS_BN 1e-5f

// ======================= FPS =======================
// One block per batch element. dist[] kept in LDS; deterministic tree argmax
// with first-index tie-break (matches jnp.argmax).
__global__ void fps_kernel(const float* __restrict__ xyz, int N, int npoint,
                           int* __restrict__ out) {
  __shared__ float dist[8192];
  __shared__ float rv[1024];
  __shared__ int   ri[1024];
  const int b = blockIdx.x, t = threadIdx.x, T = blockDim.x;
  const float* X = xyz + (size_t)b * 3 * N;
  for (int i = t; i < N; i += T) dist[i] = 1e10f;
  __syncthreads();
  int far = 0;
  for (int it = 0; it < npoint; ++it) {
    if (t == 0) out[b * npoint + it] = far;
    const float fx = X[far], fy = X[N + far], fz = X[2 * N + far];
    float best = -1.0f; int bi = 0;
    for (int i = t; i < N; i += T) {
      float dx = X[i] - fx, dy = X[N + i] - fy, dz = X[2 * N + i] - fz;
      float d = dx * dx + dy * dy + dz * dz;
      float dm = fminf(dist[i], d);
      dist[i] = dm;
      if (dm > best) { best = dm; bi = i; }
    }
    rv[t] = best; ri[t] = bi;
    __syncthreads();
    for (int s = T >> 1; s > 0; s >>= 1) {
      if (t < s) {
        if (rv[t + s] > rv[t] || (rv[t + s] == rv[t] && ri[t + s] < ri[t])) {
          rv[t] = rv[t + s]; ri[t] = ri[t + s];
        }
      }
      __syncthreads();
    }
    far = ri[0];
    __syncthreads();
  }
}

// gather new_xyz (B,3,np) from fps indices
__global__ void gather_newxyz_kernel(const float* __restrict__ xyz,
                                     const int* __restrict__ fpsidx,
                                     int N, int np, float* __restrict__ nxyz) {
  int g = blockIdx.x * blockDim.x + threadIdx.x;
  if (g >= BATCH * np) return;
  int b = g / np, s = g % np;
  int j = fpsidx[b * np + s];
  for (int c = 0; c < 3; ++c)
    nxyz[((size_t)b * 3 + c) * np + s] = xyz[((size_t)b * 3 + c) * N + j];
}

// ======================= Ball query =======================
// Sequential index scan == sort ascending, truncate, pad-with-first.
__global__ void ballquery_kernel(const float* __restrict__ xyz,
                                 const float* __restrict__ nxyz,
                                 int N, int np, float r2,
                                 int* __restrict__ idx) {
  int g = blockIdx.x * blockDim.x + threadIdx.x;
  if (g >= BATCH * np) return;
  int b = g / np, s = g % np;
  const float* X = xyz + (size_t)b * 3 * N;
  float qx = nxyz[((size_t)b * 3 + 0) * np + s];
  float qy = nxyz[((size_t)b * 3 + 1) * np + s];
  float qz = nxyz[((size_t)b * 3 + 2) * np + s];
  int base = (b * np + s) * NSAMP;
  int cnt = 0, first = 0;
  for (int i = 0; i < N && cnt < NSAMP; ++i) {
    float dx = X[i] - qx, dy = X[N + i] - qy, dz = X[2 * N + i] - qz;
    float d = dx * dx + dy * dy + dz * dz;
    if (d <= r2) {
      if (cnt == 0) first = i;
      idx[base + cnt] = i;
      ++cnt;
    }
  }
  for (int k = cnt; k < NSAMP; ++k) idx[base + k] = first;
}

// ======================= Grouping =======================
// feat layout (Cpad, M) with M = B*np*ns, m = ((b*np+s)*ns+k).
// Rows [3+Cp, Cpad) are zero-filled so the GEMM K-loop needs no guards.
__global__ void group_kernel(const float* __restrict__ xyz,
                             const float* __restrict__ pts,
                             const float* __restrict__ nxyz,
                             const int* __restrict__ idx,
                             int N, int np, int Cp, int Cpad,
                             float* __restrict__ feat) {
  int m = blockIdx.x * blockDim.x + threadIdx.x;
  const int M = BATCH * np * NSAMP;
  if (m >= M) return;
  int b = m / (np * NSAMP);
  int r = m % (np * NSAMP);
  int s = r / NSAMP;
  int j = idx[m];
  for (int c = 0; c < 3; ++c)
    feat[(size_t)c * M + m] = xyz[((size_t)b * 3 + c) * N + j] -
                              nxyz[((size_t)b * 3 + c) * np + s];
  for (int c = 0; c < Cp; ++c)
    feat[(size_t)(3 + c) * M + m] = pts[((size_t)b * Cp + c) * N + j];
  for (int c = 3 + Cp; c < Cpad; ++c)
    feat[(size_t)c * M + m] = 0.0f;
}

// ======================= Weight repack =======================
// W (cout,cin) row-major -> zero-padded (cout,cinp), cinp = ceil4(cin).
__global__ void repack_w_kernel(const float* __restrict__ W, int cin, int cinp,
                                int cout, float* __restrict__ Wp) {
  int g = blockIdx.x * blockDim.x + threadIdx.x;
  if (g >= cout * cinp) return;
  int o = g / cinp, k = g % cinp;
  Wp[g] = (k < cin) ? W[(size_t)o * cin + k] : 0.0f;
}

// ======================= WMMA f32 GEMM =======================
// Y(cout,M) = Wp(cout,cinp) * X(cinp,M) + bias.  Each wave computes a
// 32(cout) x 64(M) tile: 8 accumulators, K stepped by 4 with
// V_WMMA_F32_16X16X4_F32.  No guards in the K loop (K padded), pure
// pointer-increment addressing.
// A (16x4): lane%16 = M row; VGPR0/1 = K {0,1} (lanes 0-15) / {2,3} (16-31).
// B (4x16): lane%16 = N col; same K-half split.  C/D: VGPR r = row r / r+8.
typedef union { v8f v; float f[8]; } v8fu;

__global__ __launch_bounds__(256) void gemm_wmma_kernel(
    const float* __restrict__ X, const float* __restrict__ Wp,
    const float* __restrict__ bias, float* __restrict__ Y,
    int cinp, int cout, int M) {
  const int lane = threadIdx.x & 31;
  const int wave = threadIdx.x >> 5;
  const int half = lane >> 4;
  const int l16  = lane & 15;
  const int tilesM = M >> 6;           // M is always a multiple of 64 here
  const int tilesO = cout >> 5;        // cout always a multiple of 32
  int tid = blockIdx.x * 8 + wave;
  if (tid >= tilesO * tilesM) return;
  const int o0 = (tid / tilesM) << 5;
  const int m0 = (tid % tilesM) << 6;

  v8fu acc[2][4];
#pragma unroll
  for (int u = 0; u < 2; ++u)
#pragma unroll
    for (int t = 0; t < 4; ++t)
#pragma unroll
      for (int r = 0; r < 8; ++r)
        acc[u][t].f[r] = bias[o0 + u * 16 + r + half * 8];

  // A pointers: row (o0 [+16] + l16), column half*2; aligned 8B (cinp % 4 == 0)
  const float* wp0 = Wp + (size_t)(o0 + l16) * cinp + half * 2;
  const float* wp1 = wp0 + (size_t)16 * cinp;
  // B pointers: rows half*2 and half*2+1, column m0 + l16
  const float* xp0 = X + (size_t)(half * 2) * M + m0 + l16;
  const float* xp1 = xp0 + M;
  const size_t xstep = (size_t)4 * M;

  for (int k0 = 0; k0 < cinp; k0 += 4) {
    v2f a0 = *(const v2f*)wp0;
    v2f a1 = *(const v2f*)wp1;
    v2f bb[4];
#pragma unroll
    for (int t = 0; t < 4; ++t) {
      v2f b = {xp0[t * 16], xp1[t * 16]};
      bb[t] = b;
    }
#pragma unroll
    for (int t = 0; t < 4; ++t)
      acc[0][t].v = __builtin_amdgcn_wmma_f32_16x16x4_f32(
          false, a0, false, bb[t], (short)0, acc[0][t].v, false, false);
#pragma unroll
    for (int t = 0; t < 4; ++t)
      acc[1][t].v = __builtin_amdgcn_wmma_f32_16x16x4_f32(
          false, a1, false, bb[t], (short)0, acc[1][t].v, false, false);
    wp0 += 4; wp1 += 4; xp0 += xstep; xp1 += xstep;
  }

#pragma unroll
  for (int u = 0; u < 2; ++u)
#pragma unroll
    for (int t = 0; t < 4; ++t) {
      const int col = m0 + t * 16 + l16;
#pragma unroll
      for (int r = 0; r < 8; ++r)
        Y[(size_t)(o0 + u * 16 + r + half * 8) * M + col] = acc[u][t].f[r];
    }
}

// ======================= BN statistics (deterministic) =======================
__global__ void reduce_kernel(const float* __restrict__ Y, int M,
                              float* __restrict__ sums, int cout) {
  __shared__ float s1[256], s2[256];
  const int o = blockIdx.x, t = threadIdx.x;
  float a = 0.0f, q = 0.0f;
  for (int m = t; m < M; m += 256) {
    float v = Y[(size_t)o * M + m];
    a += v; q += v * v;
  }
  s1[t] = a; s2[t] = q;
  __syncthreads();
  for (int s = 128; s > 0; s >>= 1) {
    if (t < s) { s1[t] += s1[t + s]; s2[t] += s2[t + s]; }
    __syncthreads();
  }
  if (t == 0) { sums[o] = s1[0]; sums[cout + o] = s2[0]; }
}

__global__ void bn_relu_kernel(float* __restrict__ Y,
                               const float* __restrict__ sums,
                               const float* __restrict__ gam,
                               const float* __restrict__ bet,
                               int cout, int M) {
  size_t i = (size_t)blockIdx.x * blockDim.x + threadIdx.x;
  if (i >= (size_t)cout * M) return;
  int o = (int)(i / M);
  float mean = sums[o] / (float)M;
  float var  = sums[cout + o] / (float)M - mean * mean;
  float v = Y[i];
  v = gam[o] * (v - mean) * rsqrtf(var + EPS_BN) + bet[o];
  Y[i] = fmaxf(v, 0.0f);
}

// ======================= Max pool over nsample =======================
__global__ void maxpool_kernel(const float* __restrict__ Y, int C, int np,
                               float* __restrict__ out) {
  int g = blockIdx.x * blockDim.x + threadIdx.x;
  if (g >= BATCH * C * np) return;
  int b = g / (C * np);
  int o = (g / np) % C;
  int s = g % np;
  const int M = BATCH * np * NSAMP;
  size_t base = (size_t)o * M + (size_t)(b * np + s) * NSAMP;
  float m = -1e30f;
  for (int k = 0; k < NSAMP; ++k) m = fmaxf(m, Y[base + k]);
  out[((size_t)b * C + o) * np + s] = m;
}

// ======================= FP interpolation + concat =======================
// feat layout (C1+C2, B*N1); rows [0,C1) copied from p1, rows [C1,C1+C2)
// are 3-NN inverse-distance interpolation from p2.  (C1+C2 is always a
// multiple of 4 for the FP stages, so no extra padding rows needed.)
__global__ void fp_interp_kernel(const float* __restrict__ x1,
                                 const float* __restrict__ x2,
                                 const float* __restrict__ p1,
                                 const float* __restrict__ p2,
                                 int N1, int N2, int C1, int C2,
                                 float* __restrict__ feat) {
  int g = blockIdx.x * blockDim.x + threadIdx.x;
  if (g >= BATCH * N1) return;
  int b = g / N1, n = g % N1;
  const int M = BATCH * N1;
  float qx = x1[((size_t)b * 3 + 0) * N1 + n];
  float qy = x1[((size_t)b * 3 + 1) * N1 + n];
  float qz = x1[((size_t)b * 3 + 2) * N1 + n];
  const float* X2 = x2 + (size_t)b * 3 * N2;
  float d0 = 1e30f, d1 = 1e30f, d2 = 1e30f;
  int i0 = 0, i1 = 0, i2 = 0;
  for (int j = 0; j < N2; ++j) {
    float dx = X2[j] - qx, dy = X2[N2 + j] - qy, dz = X2[2 * N2 + j] - qz;
    float d = dx * dx + dy * dy + dz * dz;
    if (d < d0)      { d2 = d1; i2 = i1; d1 = d0; i1 = i0; d0 = d; i0 = j; }
    else if (d < d1) { d2 = d1; i2 = i1; d1 = d;  i1 = j; }
    else if (d < d2) { d2 = d;  i2 = j; }
  }
  float r0 = 1.0f / (d0 + 1e-8f);
  float r1 = 1.0f / (d1 + 1e-8f);
  float r2 = 1.0f / (d2 + 1e-8f);
  float inv = 1.0f / (r0 + r1 + r2);
  float w0 = r0 * inv, w1 = r1 * inv, w2 = r2 * inv;
  for (int c = 0; c < C1; ++c)
    feat[(size_t)c * M + g] = p1[((size_t)b * C1 + c) * N1 + n];
  for (int c = 0; c < C2; ++c) {
    const float* row = p2 + ((size_t)b * C2 + c) * N2;
    feat[(size_t)(C1 + c) * M + g] = w0 * row[i0] + w1 * row[i1] + w2 * row[i2];
  }
}

// (C, B*N) channel-major  ->  (B, C, N)
__global__ void transpose_kernel(const float* __restrict__ Y, int C, int Nn,
                                 float* __restrict__ out) {
  int g = blockIdx.x * blockDim.x + threadIdx.x;
  if (g >= BATCH * C * Nn) return;
  int b = g / (C * Nn);
  int c = (g / Nn) % C;
  int n = g % Nn;
  out[g] = Y[(size_t)c * (BATCH * Nn) + (size_t)b * Nn + n];
}

// extract first 3 channels of input (B,15,N) -> (B,3,N)
__global__ void extract_xyz_kernel(const float* __restrict__ pts, int N,
                                   float* __restrict__ xyz) {
  int g = blockIdx.x * blockDim.x + threadIdx.x;
  if (g >= BATCH * 3 * N) return;
  int b = g / (3 * N);
  int c = (g / N) % 3;
  int n = g % N;
  xyz[g] = pts[((size_t)b * 15 + c) * N + n];
}

// ======================= Host orchestration =======================
struct LayerCfg { int cin, cout; };
static const LayerCfg LCFG[21] = {
  {18, 32}, {32, 32}, {32, 64},        // sa1
  {67, 64}, {64, 64}, {64, 128},       // sa2
  {131, 128}, {128, 128}, {128, 256},  // sa3
  {259, 256}, {256, 256}, {256, 512},  // sa4
  {768, 256}, {256, 256},              // fp4
  {384, 256}, {256, 256},              // fp3
  {320, 256}, {256, 128},              // fp2
  {128, 128}, {128, 128}, {128, 128}   // fp1
};

static inline int pad4(int x) { return (x + 3) & ~3; }

static inline void run_layer(const float* X, float* Y, int li, int M,
                             void* const* d_in, float* sums, float* wpad,
                             hipStream_t stream) {
  const int cin = LCFG[li].cin, cout = LCFG[li].cout;
  const int cinp = pad4(cin);
  const float* W  = (const float*)d_in[1 + li * 4 + 0];
  const float* bb = (const float*)d_in[1 + li * 4 + 1];
  const float* gg = (const float*)d_in[1 + li * 4 + 2];
  const float* be = (const float*)d_in[1 + li * 4 + 3];
  int wn = cout * cinp;
  repack_w_kernel<<<(wn + 255) / 256, 256, 0, stream>>>(W, cin, cinp, cout, wpad);
  int waves = (cout >> 5) * (M >> 6);
  gemm_wmma_kernel<<<(waves + 7) / 8, 256, 0, stream>>>(X, wpad, bb, Y, cinp, cout, M);
  reduce_kernel<<<cout, 256, 0, stream>>>(Y, M, sums, cout);
  size_t tot = (size_t)cout * M;
  bn_relu_kernel<<<(int)((tot + 255) / 256), 256, 0, stream>>>(Y, sums, gg, be, cout, M);
}

static inline void sa_stage(const float* xyz_in, const float* pts_in, int N,
                            int np, float radius, int Cp, int li0,
                            float* xyz_out, float* p_out, int Cout,
                            float* bufA, float* bufB, float* sums, float* wpad,
                            int* fpsidx, int* ballidx, void* const* d_in,
                            hipStream_t stream) {
  fps_kernel<<<BATCH, 1024, 0, stream>>>(xyz_in, N, np, fpsidx);
  int q = BATCH * np;
  gather_newxyz_kernel<<<(q + 255) / 256, 256, 0, stream>>>(xyz_in, fpsidx, N, np, xyz_out);
  ballquery_kernel<<<(q + 255) / 256, 256, 0, stream>>>(xyz_in, xyz_out, N, np,
                                                        radius * radius, ballidx);
  int M = BATCH * np * NSAMP;
  int Cpad = pad4(3 + Cp);
  group_kernel<<<(M + 255) / 256, 256, 0, stream>>>(xyz_in, pts_in, xyz_out,
                                                    ballidx, N, np, Cp, Cpad, bufA);
  run_layer(bufA, bufB, li0 + 0, M, d_in, sums, wpad, stream);
  run_layer(bufB, bufA, li0 + 1, M, d_in, sums, wpad, stream);
  run_layer(bufA, bufB, li0 + 2, M, d_in, sums, wpad, stream);
  int mp = BATCH * Cout * np;
  maxpool_kernel<<<(mp + 255) / 256, 256, 0, stream>>>(bufB, Cout, np, p_out);
}

static inline void fp_stage(const float* x1, const float* x2, const float* p1,
                            const float* p2, int N1, int N2, int C1, int C2,
                            int li0, int nlayers, float* out,
                            float* bufA, float* bufB, float* sums, float* wpad,
                            void* const* d_in, hipStream_t stream) {
  int q = BATCH * N1;
  fp_interp_kernel<<<(q + 255) / 256, 256, 0, stream>>>(x1, x2, p1, p2, N1, N2,
                                                        C1, C2, bufA);
  int M = BATCH * N1;
  run_layer(bufA, bufB, li0 + 0, M, d_in, sums, wpad, stream);
  run_layer(bufB, bufA, li0 + 1, M, d_in, sums, wpad, stream);
  const float* src = bufA;
  int Cfin = LCFG[li0 + nlayers - 1].cout;
  if (nlayers == 3) {
    run_layer(bufA, bufB, li0 + 2, M, d_in, sums, wpad, stream);
    src = bufB;
  }
  int tt = BATCH * Cfin * N1;
  transpose_kernel<<<(tt + 255) / 256, 256, 0, stream>>>(src, Cfin, N1, out);
}

extern "C" void kernel_launch(void* const* d_in, const int* in_sizes, int n_in,
                              void* d_out, int out_size, void* d_ws, size_t ws_size,
                              hipStream_t stream) {
  const float* pts = (const float*)d_in[0];   // (8,15,8192)

  // Workspace layout (floats)
  float* bufA  = (float*)d_ws;                    // 16,777,216 (64x262144)
  float* bufB  = bufA  + 16777216;                // 16,777,216
  float* sa1_p = bufB  + 16777216;                // 8*64*1024
  float* sa2_p = sa1_p + 524288;                  // 8*128*256
  float* sa3_p = sa2_p + 262144;                  // 8*256*64
  float* sa4_p = sa3_p + 131072;                  // 8*512*16
  float* fp4_p = sa4_p + 65536;                   // 8*256*64
  float* fp3_p = fp4_p + 131072;                  // 8*256*256
  float* fp2_p = fp3_p + 524288;                  // 8*128*1024
  float* l0x   = fp2_p + 1048576;                 // 8*3*8192
  float* l1x   = l0x   + 196608;                  // 8*3*1024
  float* l2x   = l1x   + 24576;                   // 8*3*256
  float* l3x   = l2x   + 6144;                    // 8*3*64
  float* l4x   = l3x   + 1536;                    // 8*3*16
  float* sums  = l4x   + 384;                     // 1024
  float* wpad  = sums  + 1024;                    // 262144 (max 768*256 weights)
  int* fpsidx  = (int*)(wpad + 262144);           // 8*1024
  int* ballidx = fpsidx + 8192;                   // 8*1024*32

  int e0 = BATCH * 3 * 8192;
  extract_xyz_kernel<<<(e0 + 255) / 256, 256, 0, stream>>>(pts, 8192, l0x);

  // Set abstraction
  sa_stage(l0x, pts,   8192, 1024, 0.1f, 15,  0, l1x, sa1_p,  64, bufA, bufB, sums, wpad, fpsidx, ballidx, d_in, stream);
  sa_stage(l1x, sa1_p, 1024,  256, 0.2f, 64,  3, l2x, sa2_p, 128, bufA, bufB, sums, wpad, fpsidx, ballidx, d_in, stream);
  sa_stage(l2x, sa2_p,  256,   64, 0.4f, 128, 6, l3x, sa3_p, 256, bufA, bufB, sums, wpad, fpsidx, ballidx, d_in, stream);
  sa_stage(l3x, sa3_p,   64,   16, 0.8f, 256, 9, l4x, sa4_p, 512, bufA, bufB, sums, wpad, fpsidx, ballidx, d_in, stream);

  // Feature propagation
  fp_stage(l3x, l4x, sa3_p, sa4_p,   64,   16, 256, 512, 12, 2, fp4_p, bufA, bufB, sums, wpad, d_in, stream);
  fp_stage(l2x, l3x, sa2_p, fp4_p,  256,   64, 128, 256, 14, 2, fp3_p, bufA, bufB, sums, wpad, d_in, stream);
  fp_stage(l1x, l2x, sa1_p, fp3_p, 1024,  256,  64, 256, 16, 2, fp2_p, bufA, bufB, sums, wpad, d_in, stream);
  fp_stage(l0x, l1x, nullptr, fp2_p, 8192, 1024, 0, 128, 18, 3, (float*)d_out, bufA, bufB, sums, wpad, d_in, stream);
}